// OuputHeadDet_59854664237218
// MI455X (gfx1250) — compile-verified
//
#include <hip/hip_runtime.h>
#include <hip/hip_bf16.h>

// Problem constants (fixed by the reference)
#define B_WALKERS 4096
#define NSLOT     64      // determinant slots per spin; orbitals has 2*NSLOT rows
#define DMODEL    256
#define NELEC     16

typedef __attribute__((ext_vector_type(2))) float v2f;
typedef __attribute__((ext_vector_type(8))) float v8f;

// ---------------------------------------------------------------------------
// Kernel 1: orbitals[b, s, e] for s in [0, 128)   (s<64: up, s>=64: down)
// One wave computes one 16(walkers) x 16(electrons) tile for one slot,
// accumulating BOTH spins so each y row is read exactly once from HBM.
// Uses V_WMMA_F32_16X16X4_F32 (full f32 precision), K-loop over D=256.
// ---------------------------------------------------------------------------
__global__ __launch_bounds__(128) void orbitals_wmma_kernel(
    const float* __restrict__ y,    // (B, NSLOT, DMODEL)
    const float* __restrict__ Mu,   // (NSLOT, DMODEL, NELEC)
    const float* __restrict__ Md,   // (NSLOT, DMODEL, NELEC)
    float* __restrict__ orb)        // (B, 2*NSLOT, NELEC)
{
    const int lane = threadIdx.x & 31;
    const int wave = threadIdx.x >> 5;
    const int tile = blockIdx.x * 4 + wave;   // 0..255 walker tile (16 walkers)
    const int slot = blockIdx.y;              // 0..63
    const int b0   = tile * 16;
    const int m    = lane & 15;               // row-in-tile (A) / column (B, C/D)
    const int h    = lane >> 4;               // lane half: selects K pair

    // A operand: lane (m, h) supplies y[b0+m, slot, k0] and [k0+1], k0 = d0+2h
    const float* yrow = y  + ((size_t)(b0 + m) * NSLOT + slot) * DMODEL;
    const float* mu   = Mu + (size_t)slot * DMODEL * NELEC;
    const float* md   = Md + (size_t)slot * DMODEL * NELEC;

    v8f accu = {};
    v8f accd = {};

#pragma unroll 4
    for (int d0 = 0; d0 < DMODEL; d0 += 4) {
        const int k0 = d0 + 2 * h;
        v2f a;                                    // A 16x4: K=k0, k0+1 per lane
        a.x = yrow[k0];
        a.y = yrow[k0 + 1];
        v2f bu, bd;                               // B 4x16: rows k0, k0+1, col m
        bu.x = mu[(size_t)k0 * NELEC + m];
        bu.y = mu[(size_t)(k0 + 1) * NELEC + m];
        bd.x = md[(size_t)k0 * NELEC + m];
        bd.y = md[(size_t)(k0 + 1) * NELEC + m];
        // 8 args: (neg_a, A, neg_b, B, c_mod, C, reuse_a, reuse_b)
        accu = __builtin_amdgcn_wmma_f32_16x16x4_f32(
            false, a, false, bu, (short)0, accu, false, false);
        accd = __builtin_amdgcn_wmma_f32_16x16x4_f32(
            false, a, false, bd, (short)0, accd, false, false);
    }

    // C/D layout: VGPR v -> tile row (v + 8h), column m
    const size_t wstride = (size_t)2 * NSLOT * NELEC;   // walker stride in orb
    float* oru = orb + ((size_t)b0 * 2 * NSLOT + slot) * NELEC;
    float* ord = orb + ((size_t)b0 * 2 * NSLOT + (slot + NSLOT)) * NELEC;
#pragma unroll
    for (int v = 0; v < 8; ++v) {
        const int row = v + 8 * h;
        oru[(size_t)row * wstride + m] = accu[v];
        ord[(size_t)row * wstride + m] = accd[v];
    }
}

// ---------------------------------------------------------------------------
// Kernel 2: gather 16 rows per walker via R, LU with partial pivoting in LDS,
// emit complex64 amp = logabsdet + log(sign); NaN -> -inf + 0j.
// LDS layout lds[elem*32 + tid]: bank == tid, conflict-free under divergence.
// ---------------------------------------------------------------------------
__global__ __launch_bounds__(32) void slogdet_kernel(
    const float* __restrict__ orb,  // (B, 2*NSLOT, NELEC)
    const int*   __restrict__ R,    // (B, NELEC)
    float*       __restrict__ out)  // (B) complex64 -> 2*B floats interleaved
{
    __shared__ float lds[NELEC * NELEC * 32];   // 32 KB
    const int t = threadIdx.x;
    const int b = blockIdx.x * 32 + t;

#define AEL(i, j) lds[((((i) << 4) + (j)) << 5) + t]

    // Gather A[i,:] = orbitals[b, R[b,i], :]
    const int*   r  = R + (size_t)b * NELEC;
    const float* ob = orb + (size_t)b * 2 * NSLOT * NELEC;
    for (int i = 0; i < NELEC; ++i) {
        const float* src = ob + (size_t)r[i] * NELEC;
#pragma unroll
        for (int j = 0; j < NELEC; ++j) AEL(i, j) = src[j];
    }

    float sgn = 1.0f;
    float logabs = 0.0f;
    for (int k = 0; k < NELEC; ++k) {
        // partial pivot search in column k
        int p = k;
        float mv = fabsf(AEL(k, k));
        for (int i = k + 1; i < NELEC; ++i) {
            float v = fabsf(AEL(i, k));
            if (v > mv) { mv = v; p = i; }
        }
        if (p != k) {
            sgn = -sgn;
            for (int j = 0; j < NELEC; ++j) {
                float tmp = AEL(k, j); AEL(k, j) = AEL(p, j); AEL(p, j) = tmp;
            }
        }
        const float piv = AEL(k, k);
        sgn *= (piv < 0.0f) ? -1.0f : ((piv > 0.0f) ? 1.0f : 0.0f);
        logabs += logf(fabsf(piv));          // piv==0 -> -inf (matches slogdet)
        const float inv = 1.0f / piv;
        for (int i = k + 1; i < NELEC; ++i) {
            const float f = AEL(i, k) * inv;
            for (int j = k + 1; j < NELEC; ++j) AEL(i, j) -= f * AEL(k, j);
        }
    }
#undef AEL

    float re = logabs;
    float im = (sgn < 0.0f) ? 3.14159265358979323846f : 0.0f;
    // jnp.where(isnan(amp), -inf, amp)
    if (!(re == re) || !(im == im)) { re = -__builtin_inff(); im = 0.0f; }
    out[2 * b]     = re;
    out[2 * b + 1] = im;
}

// ---------------------------------------------------------------------------
extern "C" void kernel_launch(void* const* d_in, const int* in_sizes, int n_in,
                              void* d_out, int out_size, void* d_ws, size_t ws_size,
                              hipStream_t stream) {
    const float* y  = (const float*)d_in[0];
    const float* Mu = (const float*)d_in[1];
    const float* Md = (const float*)d_in[2];
    const int*   R  = (const int*)d_in[3];
    float* out = (float*)d_out;
    float* orb = (float*)d_ws;   // needs B * 128 * 16 * 4 = 32 MB scratch

    // Kernel 1: grid (256 walker-tiles / 4 waves-per-block, 64 slots)
    dim3 g1(B_WALKERS / 16 / 4, NSLOT, 1);
    orbitals_wmma_kernel<<<g1, 128, 0, stream>>>(y, Mu, Md, orb);

    // Kernel 2: one wave32 block handles 32 walkers
    slogdet_kernel<<<B_WALKERS / 32, 32, 0, stream>>>(orb, R, out);
}